// Attention_25512105738292
// MI455X (gfx1250) — compile-verified
//
#include <hip/hip_runtime.h>

// Bahdanau additive attention, MI455X (gfx1250, wave32).
// T=2048, B=32, N=1024, fp32. Memory-bound: 512 MB streamed => ~22us floor @23.3TB/s.
// - Skinny GEMMs use native fp32 WMMA: V_WMMA_F32_16X16X4_F32.
// - Streaming loads use nontemporal (TH=NT) to keep reused data resident in L2.
// - tanh uses CDNA5 native V_TANH_F32 when available.

typedef __attribute__((ext_vector_type(2))) float v2f;
typedef __attribute__((ext_vector_type(4))) float v4f;
typedef __attribute__((ext_vector_type(8))) float v8f;

#define T_DIM 2048
#define B_DIM 32
#define N_DIM 1024

__device__ __forceinline__ float fast_tanh(float x) {
#if __has_builtin(__builtin_amdgcn_tanhf)
  return __builtin_amdgcn_tanhf(x);   // V_TANH_F32 (single trans op)
#else
  return tanhf(x);
#endif
}

__device__ __forceinline__ v4f nt_load4(const float* p) {
  return __builtin_nontemporal_load((const v4f*)p);  // global_load_b128 th:NT
}

// ---------------------------------------------------------------------------
// K1: dec_feature[m][n] = sum_k hx[m][k] * Ws_w[n][k] + Ws_b[n]   (32 x 1024)
// One wave per 16x16 output tile; K-loop in steps of 4 via V_WMMA_F32_16X16X4_F32.
// A layout: lanes 0-15 -> M=0..15 K={k,k+1}; lanes 16-31 -> M=0..15 K={k+2,k+3}.
// B layout: lane%16 -> N col; VGPR0/1 x lane-half -> K rows (mirrors A striping).
// C/D: VGPR r, lane L -> row r + 8*(L>=16), col L%16.
// ---------------------------------------------------------------------------
__global__ __launch_bounds__(32) void k_dec_feature(const float* __restrict__ hx,
                                                    const float* __restrict__ Ws_w,
                                                    const float* __restrict__ Ws_b,
                                                    float* __restrict__ dec) {
  const int lane = threadIdx.x;
  const int half = lane >> 4;  // 0 or 1
  const int lm   = lane & 15;
  const int mtile = blockIdx.y;   // 0..1
  const int ntile = blockIdx.x;   // 0..63
  const int m = mtile * 16 + lm;  // A row for this lane
  const int n = ntile * 16 + lm;  // B col == D col for this lane

  v8f c = {};
  for (int k = 0; k < N_DIM; k += 4) {
    const int kk = k + half * 2;
    v2f a = *(const v2f*)(hx   + (size_t)m * N_DIM + kk);   // hx[m][kk..kk+1]
    v2f b = *(const v2f*)(Ws_w + (size_t)n * N_DIM + kk);   // Ws_w[n][kk..kk+1]
    c = __builtin_amdgcn_wmma_f32_16x16x4_f32(false, a, false, b, (short)0, c,
                                              false, false);
  }
  const float bias = Ws_b[n];
#pragma unroll
  for (int r = 0; r < 8; ++r) {
    const int row = mtile * 16 + half * 8 + r;
    dec[(size_t)row * N_DIM + n] = c[r] + bias;
  }
}

// ---------------------------------------------------------------------------
// K2: scores[b][t] = sum_n tanh(enc_feat[t][b][n] + dec[b][n]) * v_w[n] + v_b
// One wave per (b, 8 consecutive t): dec row + v_w are loaded into registers
// ONCE and amortized over 8 streamed 4KB rows (cuts L2 request traffic ~3x).
// enc_feat is nontemporal (read once, 256MB). 8192 waves in flight class.
// ---------------------------------------------------------------------------
__global__ __launch_bounds__(256) void k_scores(const float* __restrict__ enc_feat,
                                                const float* __restrict__ dec,
                                                const float* __restrict__ v_w,
                                                const float* __restrict__ v_b,
                                                float* __restrict__ scores) {
  const int lane = threadIdx.x & 31;
  const int p    = blockIdx.x * (blockDim.x >> 5) + (threadIdx.x >> 5);  // wave id
  const int b  = p & (B_DIM - 1);
  const int t0 = (p >> 5) * 8;   // 8 t's per wave

  const float* dr = dec + (size_t)b * N_DIM + lane * 4;
  const float* vr = v_w + lane * 4;
  v4f d[8], v[8];
#pragma unroll
  for (int i = 0; i < 8; ++i) {
    d[i] = *(const v4f*)(dr + i * 128);
    v[i] = *(const v4f*)(vr + i * 128);
  }
  const float vb = v_b[0];

  for (int j = 0; j < 8; ++j) {
    const int t = t0 + j;
    const float* ef = enc_feat + ((size_t)t * B_DIM + b) * N_DIM + lane * 4;
    float acc = 0.f;
#pragma unroll
    for (int i = 0; i < 8; ++i) {
      const v4f e = nt_load4(ef + i * 128);
      acc += fast_tanh(e.x + d[i].x) * v[i].x;
      acc += fast_tanh(e.y + d[i].y) * v[i].y;
      acc += fast_tanh(e.z + d[i].z) * v[i].z;
      acc += fast_tanh(e.w + d[i].w) * v[i].w;
    }
#pragma unroll
    for (int off = 16; off; off >>= 1) acc += __shfl_xor(acc, off, 32);
    if (lane == 0) scores[(size_t)b * T_DIM + t] = acc + vb;
  }
}

// ---------------------------------------------------------------------------
// K3: in-place masked softmax over t for each b (32 rows x 2048).
// ---------------------------------------------------------------------------
__global__ __launch_bounds__(256) void k_softmax(float* __restrict__ scores,
                                                 const float* __restrict__ mask) {
  const int b = blockIdx.x;
  float* row = scores + (size_t)b * T_DIM;
  const int tid = threadIdx.x;
  __shared__ float sred[8];

  float m = -3.4028235e38f;
  for (int t = tid; t < T_DIM; t += 256) m = fmaxf(m, row[t]);
#pragma unroll
  for (int off = 16; off; off >>= 1) m = fmaxf(m, __shfl_xor(m, off, 32));
  if ((tid & 31) == 0) sred[tid >> 5] = m;
  __syncthreads();
  if (tid == 0) {
    float mm = sred[0];
    for (int i = 1; i < 8; ++i) mm = fmaxf(mm, sred[i]);
    sred[0] = mm;
  }
  __syncthreads();
  m = sred[0];
  __syncthreads();

  float sum = 0.f;
  for (int t = tid; t < T_DIM; t += 256) sum += __expf(row[t] - m);
#pragma unroll
  for (int off = 16; off; off >>= 1) sum += __shfl_xor(sum, off, 32);
  if ((tid & 31) == 0) sred[tid >> 5] = sum;
  __syncthreads();
  if (tid == 0) {
    float s = 0.f;
    for (int i = 0; i < 8; ++i) s += sred[i];
    sred[0] = s;
  }
  __syncthreads();
  const float inv = 1.0f / sred[0];
  for (int t = tid; t < T_DIM; t += 256)
    row[t] = __expf(row[t] - m) * inv * mask[(size_t)b * T_DIM + t];
}

// ---------------------------------------------------------------------------
// K4: content[b][n] += sum_{t in chunk} attn[b][t] * enc_out[t][b][n]
// grid (B, 32 t-chunks) x 256 threads; each thread owns 4 consecutive n.
// Each iteration the block streams one contiguous 4KB row (nontemporal).
// Cross-chunk combine via global fp32 atomics (content memset to 0 first).
// ---------------------------------------------------------------------------
__global__ __launch_bounds__(256) void k_context(const float* __restrict__ enc_out,
                                                 const float* __restrict__ attn,
                                                 float* __restrict__ content) {
  const int b  = blockIdx.x;
  const int tc = blockIdx.y;
  const int n4 = threadIdx.x * 4;
  const int t0 = tc * (T_DIM / 32);
  const int t1 = t0 + (T_DIM / 32);

  v4f acc = {0.f, 0.f, 0.f, 0.f};
  for (int t = t0; t < t1; ++t) {
    if (t + 16 < t1)  // uniform branch; prefetch ahead (global_prefetch)
      __builtin_prefetch(enc_out + ((size_t)(t + 16) * B_DIM + b) * N_DIM + n4, 0, 1);
    const float w = attn[(size_t)b * T_DIM + t];
    const v4f   e = nt_load4(enc_out + ((size_t)t * B_DIM + b) * N_DIM + n4);
    acc.x += w * e.x;
    acc.y += w * e.y;
    acc.z += w * e.z;
    acc.w += w * e.w;
  }
  float* dst = content + (size_t)b * N_DIM + n4;
  atomicAdd(dst + 0, acc.x);
  atomicAdd(dst + 1, acc.y);
  atomicAdd(dst + 2, acc.z);
  atomicAdd(dst + 3, acc.w);
}

// ---------------------------------------------------------------------------
// K5: out[m][j] = tanh( sum_{k<2048} concat[m][k] * lin_w[j][k] + lin_b[j] )
// concat[m][k] = k < 1024 ? content[m][k] : hx[m][k-1024]  (uniform branch).
// Same WMMA tiling as K1, K = 2048.
// ---------------------------------------------------------------------------
__global__ __launch_bounds__(32) void k_out(const float* __restrict__ content,
                                            const float* __restrict__ hx,
                                            const float* __restrict__ lin_w,
                                            const float* __restrict__ lin_b,
                                            float* __restrict__ out) {
  const int lane = threadIdx.x;
  const int half = lane >> 4;
  const int lm   = lane & 15;
  const int mtile = blockIdx.y;   // 0..1
  const int ntile = blockIdx.x;   // 0..63
  const int m = mtile * 16 + lm;
  const int j = ntile * 16 + lm;

  v8f c = {};
  for (int k = 0; k < 2 * N_DIM; k += 4) {
    const int kk = k + half * 2;
    const float* arow = (k < N_DIM) ? (content + (size_t)m * N_DIM + kk)
                                    : (hx + (size_t)m * N_DIM + (kk - N_DIM));
    v2f a = *(const v2f*)arow;
    v2f b = *(const v2f*)(lin_w + (size_t)j * (2 * N_DIM) + kk);  // lin_w[j][kk..]
    c = __builtin_amdgcn_wmma_f32_16x16x4_f32(false, a, false, b, (short)0, c,
                                              false, false);
  }
  const float bias = lin_b[j];
#pragma unroll
  for (int r = 0; r < 8; ++r) {
    const int row = mtile * 16 + half * 8 + r;
    out[(size_t)row * N_DIM + j] = fast_tanh(c[r] + bias);
  }
}

// ---------------------------------------------------------------------------
extern "C" void kernel_launch(void* const* d_in, const int* in_sizes, int n_in,
                              void* d_out, int out_size, void* d_ws, size_t ws_size,
                              hipStream_t stream) {
  const float* hx       = (const float*)d_in[0];  // (32, 1024)
  const float* enc_out  = (const float*)d_in[1];  // (2048, 32, 1024)
  const float* enc_feat = (const float*)d_in[2];  // (2048, 32, 1024)
  const float* mask     = (const float*)d_in[3];  // (32, 2048)
  const float* Ws_w     = (const float*)d_in[4];  // (1024, 1024)
  const float* Ws_b     = (const float*)d_in[5];  // (1024,)
  const float* v_w      = (const float*)d_in[6];  // (1, 1024)
  const float* v_b      = (const float*)d_in[7];  // (1,)
  const float* lin_w    = (const float*)d_in[8];  // (1024, 2048)
  const float* lin_b    = (const float*)d_in[9];  // (1024,)
  float* out = (float*)d_out;                     // (32, 1024)

  // Workspace layout (floats): dec(32K) | scores/attn(64K) | content(32K) = 512KB
  float* ws      = (float*)d_ws;
  float* dec     = ws;
  float* scores  = ws + B_DIM * N_DIM;
  float* content = ws + B_DIM * N_DIM + B_DIM * T_DIM;

  // K1: dec_feature GEMM (WMMA fp32)
  k_dec_feature<<<dim3(N_DIM / 16, B_DIM / 16), 32, 0, stream>>>(hx, Ws_w, Ws_b, dec);

  // K2: fused tanh + rank-1 projection (streams 256MB, 8 t per wave)
  k_scores<<<(T_DIM / 8) * B_DIM / 8, 256, 0, stream>>>(enc_feat, dec, v_w, v_b, scores);

  // K3: masked softmax (in place on scores)
  k_softmax<<<B_DIM, 256, 0, stream>>>(scores, mask);

  // K4: weighted temporal sum (streams 256MB); zero accumulator first
  hipMemsetAsync(content, 0, (size_t)B_DIM * N_DIM * sizeof(float), stream);
  k_context<<<dim3(B_DIM, 32), 256, 0, stream>>>(enc_out, scores, content);

  // K5: output GEMM over virtual concat + tanh (WMMA fp32)
  k_out<<<dim3(N_DIM / 16, B_DIM / 16), 32, 0, stream>>>(content, hx, lin_w, lin_b, out);
}